// CovCorrLog_22531398434819
// MI455X (gfx1250) — compile-verified
//
#include <hip/hip_runtime.h>
#include <hip/hip_bf16.h>
#include <math.h>

// ---------------------------------------------------------------------------
// CovCorrLog on MI455X (gfx1250, wave32, WMMA)
//
// Pipeline (all batched over B=128, C=512, M=196):
//   1) center rows of x -> Xc
//   2) cov = Xc @ Xc^T / M                       (WMMA f32 GEMM, K=196)
//   3) S = sqrtm(cov) via Newton-Schulz          (12 WMMA 512^3 GEMMs)
//   4) Cm = S / (std_i std_j)                    (elementwise)
//   5) logm(Cm) via inverse scaling-and-squaring:
//        4x  X <- sqrtm_NS(X)                    (4 * 12 GEMMs)
//        R = X - I ; log ~ 2^4 * R * Horner(R)   (8 GEMMs, Gregory series)
//   6) zero diag, symmetrize, triu-vectorize -> d_out
// ---------------------------------------------------------------------------

typedef __attribute__((ext_vector_type(2))) float v2f;
typedef __attribute__((ext_vector_type(4))) float v4f;
typedef __attribute__((ext_vector_type(8))) float v8f;

#define BATCH 128
#define CDIM  512
#define MDIM  196
#define EPSF  1.1920929e-07f

// main GEMM tiling: block = 128M x 64N, 8 waves, each wave 32x32 (2x2 WMMA tiles)
#define TM 128
#define TN 64
#define TKK 32
#define PADA 36   // LDS row stride for A tile (mult of 4 -> b128 stores; conflict-free)
#define PADBT 34  // LDS row stride for K-major B tile (even -> b64 frags; conflict-free)

#define TILE_M 64
#define TILE_N 64
#define TK_COV 28   // 196 = 7 * 28, and 28 % 4 == 0

// ---------------------------------------------------------------------------
// Batched 512x512x512 GEMM:  D = alpha * (A @ B) + beta * I
// If use_sqrt_tr != 0, alpha *= sqrtf(tr[b]).
// B tile is staged K-major in LDS so both A and B fragments are single
// adjacent-pair (b64) LDS loads -- no VGPR packing movs in the WMMA loop.
// ---------------------------------------------------------------------------
__global__ __launch_bounds__(256)
void wmma_gemm_nn(const float* __restrict__ A, const float* __restrict__ B,
                  float* __restrict__ D, const float* __restrict__ tr,
                  float alpha, float beta, int use_sqrt_tr)
{
    __shared__ float lA[TM][PADA];     // 128*36*4 = 18432 B, lA[m][k]
    __shared__ float lBp[TN][PADBT];   //  64*34*4 =  8704 B, lBp[n][k] (K-major)

    const int b  = blockIdx.z;
    const int bm = blockIdx.y * TM;
    const int bn = blockIdx.x * TN;
    const float* Ab = A + ((size_t)b << 18);
    const float* Bb = B + ((size_t)b << 18);
    float*       Db = D + ((size_t)b << 18);

    const int tid  = threadIdx.x;
    const int lane = tid & 31;
    const int wid  = tid >> 5;
    const int wm   = (wid & 3) << 5;   // 0,32,64,96
    const int wn   = (wid >> 2) << 5;  // 0,32

    v8f acc00 = {}, acc01 = {}, acc10 = {}, acc11 = {};

    const int m  = lane & 15;
    const int kk = (lane >> 4) << 1;   // 0 for lanes 0-15, 2 for lanes 16-31

    // staging indices
    const int ar = tid >> 3, ac = (tid & 7) << 2;    // A: 128 rows x 8 v4f (16B aligned)
    const int bnn = tid & 63, bkp = tid >> 6;        // B: pair-transpose, 64 n x 16 kpairs

    for (int k0 = 0; k0 < CDIM; k0 += TKK) {
        // prefetch next K tile into cache while we stage/compute this one
        if (k0 + TKK < CDIM) {
            __builtin_prefetch(&Ab[(size_t)(bm + ar) * CDIM + (k0 + TKK) + ac], 0, 1);
            __builtin_prefetch(&Bb[(size_t)(k0 + TKK + bkp) * CDIM + bn + bnn], 0, 1);
        }
        // stage A tile 128x32 row-major (1024 v4f, 4 per thread)
        #pragma unroll
        for (int i = 0; i < 4; ++i) {
            int r = ar + i * 32;
            *(v4f*)&lA[r][ac] = *(const v4f*)&Ab[(size_t)(bm + r) * CDIM + k0 + ac];
        }
        // stage B tile 32x64 as K-major pairs (1024 pairs, 4 per thread)
        #pragma unroll
        for (int i = 0; i < 4; ++i) {
            int kp = bkp + i * 4;            // 0..15 -> k = 2*kp
            v2f t;
            t.x = Bb[(size_t)(k0 + 2 * kp) * CDIM + bn + bnn];
            t.y = Bb[(size_t)(k0 + 2 * kp + 1) * CDIM + bn + bnn];
            *(v2f*)&lBp[bnn][2 * kp] = t;
        }
        __syncthreads();

        #pragma unroll
        for (int k = 0; k < TKK; k += 4) {
            v2f a0 = *(const v2f*)&lA[wm + m][k + kk];
            v2f a1 = *(const v2f*)&lA[wm + 16 + m][k + kk];
            v2f b0 = *(const v2f*)&lBp[wn + m][k + kk];
            v2f b1 = *(const v2f*)&lBp[wn + 16 + m][k + kk];
            acc00 = __builtin_amdgcn_wmma_f32_16x16x4_f32(false, a0, false, b0,
                                                          (short)0, acc00, false, false);
            acc01 = __builtin_amdgcn_wmma_f32_16x16x4_f32(false, a0, false, b1,
                                                          (short)0, acc01, false, false);
            acc10 = __builtin_amdgcn_wmma_f32_16x16x4_f32(false, a1, false, b0,
                                                          (short)0, acc10, false, false);
            acc11 = __builtin_amdgcn_wmma_f32_16x16x4_f32(false, a1, false, b1,
                                                          (short)0, acc11, false, false);
        }
        __syncthreads();
    }

    float aeff = alpha;
    if (use_sqrt_tr) aeff *= sqrtf(tr[b]);

    const int mrow = (lane >> 4) << 3;  // 0 or 8
    const int n    = lane & 15;
    #pragma unroll
    for (int r = 0; r < 8; ++r) {
        int gm0 = bm + wm + mrow + r;
        int gm1 = gm0 + 16;
        int gn0 = bn + wn + n;
        int gn1 = gn0 + 16;
        Db[(size_t)gm0 * CDIM + gn0] = aeff * acc00[r] + ((gm0 == gn0) ? beta : 0.0f);
        Db[(size_t)gm0 * CDIM + gn1] = aeff * acc01[r] + ((gm0 == gn1) ? beta : 0.0f);
        Db[(size_t)gm1 * CDIM + gn0] = aeff * acc10[r] + ((gm1 == gn0) ? beta : 0.0f);
        Db[(size_t)gm1 * CDIM + gn1] = aeff * acc11[r] + ((gm1 == gn1) ? beta : 0.0f);
    }
}

// ---------------------------------------------------------------------------
// cov[b] = Xc[b] @ Xc[b]^T / M    (Xc: [B, 512, 196] row-major)
// ---------------------------------------------------------------------------
__global__ __launch_bounds__(256)
void wmma_cov_nt(const float* __restrict__ Xc, float* __restrict__ cov)
{
    __shared__ float lA[TILE_M][TK_COV + 1];
    __shared__ float lB[TK_COV][TILE_N + 1];

    const int b  = blockIdx.z;
    const int bm = blockIdx.y * TILE_M;
    const int bn = blockIdx.x * TILE_N;
    const float* Xb = Xc + (size_t)b * CDIM * MDIM;
    float*       Db = cov + ((size_t)b << 18);

    const int tid  = threadIdx.x;
    const int lane = tid & 31;
    const int wid  = tid >> 5;
    const int wm   = (wid & 3) << 4;
    const int wn   = (wid >> 2) << 5;

    v8f acc0 = {};
    v8f acc1 = {};

    const int m  = lane & 15;
    const int kk = (lane >> 4) << 1;

    for (int k0 = 0; k0 < MDIM; k0 += TK_COV) {
        #pragma unroll
        for (int i = 0; i < (TILE_M * TK_COV) / 256; ++i) {  // 1792/256 = 7
            int idx = tid + i * 256;
            int r = idx / TK_COV, c = idx % TK_COV;
            lA[r][c] = Xb[(size_t)(bm + r) * MDIM + (k0 + c)];
        }
        #pragma unroll
        for (int i = 0; i < (TK_COV * TILE_N) / 256; ++i) {
            int idx = tid + i * 256;
            int r = idx >> 6, c = idx & 63;
            lB[r][c] = Xb[(size_t)(bn + c) * MDIM + (k0 + r)];
        }
        __syncthreads();

        #pragma unroll
        for (int k = 0; k < TK_COV; k += 4) {
            v2f af;  af.x  = lA[wm + m][k + kk];      af.y  = lA[wm + m][k + kk + 1];
            v2f bf0; bf0.x = lB[k + kk][wn + m];      bf0.y = lB[k + kk + 1][wn + m];
            v2f bf1; bf1.x = lB[k + kk][wn + 16 + m]; bf1.y = lB[k + kk + 1][wn + 16 + m];
            acc0 = __builtin_amdgcn_wmma_f32_16x16x4_f32(false, af, false, bf0,
                                                         (short)0, acc0, false, false);
            acc1 = __builtin_amdgcn_wmma_f32_16x16x4_f32(false, af, false, bf1,
                                                         (short)0, acc1, false, false);
        }
        __syncthreads();
    }

    const float inv_m = 1.0f / (float)MDIM;
    const int mrow = (lane >> 4) << 3;
    const int n    = lane & 15;
    #pragma unroll
    for (int r = 0; r < 8; ++r) {
        int gm = bm + wm + mrow + r;
        Db[(size_t)gm * CDIM + (bn + wn + n)]      = acc0[r] * inv_m;
        Db[(size_t)gm * CDIM + (bn + wn + 16 + n)] = acc1[r] * inv_m;
    }
}

// ---------------------------------------------------------------------------
// small kernels
// ---------------------------------------------------------------------------
__global__ __launch_bounds__(64)
void center_k(const float* __restrict__ x, float* __restrict__ Xc)
{
    const int row = blockIdx.x;                 // b*512 + c
    const float* xr = x + (size_t)row * MDIM;
    float* yr = Xc + (size_t)row * MDIM;
    __shared__ float s[64];
    float v = 0.0f;
    for (int i = threadIdx.x; i < MDIM; i += 64) v += xr[i];
    s[threadIdx.x] = v;
    __syncthreads();
    for (int st = 32; st > 0; st >>= 1) {
        if (threadIdx.x < st) s[threadIdx.x] += s[threadIdx.x + st];
        __syncthreads();
    }
    const float mean = s[0] * (1.0f / (float)MDIM);
    for (int i = threadIdx.x; i < MDIM; i += 64) yr[i] = xr[i] - mean;
}

__global__ __launch_bounds__(256)
void trace_k(const float* __restrict__ A, float* __restrict__ tr)
{
    const int b = blockIdx.x;
    __shared__ float s[256];
    float v = 0.0f;
    for (int i = threadIdx.x; i < CDIM; i += 256)
        v += A[((size_t)b << 18) + ((size_t)i * CDIM) + i];
    s[threadIdx.x] = v;
    __syncthreads();
    for (int st = 128; st > 0; st >>= 1) {
        if (threadIdx.x < st) s[threadIdx.x] += s[threadIdx.x + st];
        __syncthreads();
    }
    if (threadIdx.x == 0) tr[b] = s[0];
}

// A = in/tr[b]; T = 1.5I - 0.5A; Z = T
__global__ __launch_bounds__(256)
void prep_ns_k(const float* __restrict__ in, const float* __restrict__ tr,
               float* __restrict__ A, float* __restrict__ T, float* __restrict__ Z,
               size_t total)
{
    for (size_t idx = (size_t)blockIdx.x * blockDim.x + threadIdx.x;
         idx < total; idx += (size_t)gridDim.x * blockDim.x) {
        int b = (int)(idx >> 18);
        int i = (int)((idx >> 9) & 511);
        int j = (int)(idx & 511);
        float a = in[idx] / tr[b];
        A[idx] = a;
        float t = ((i == j) ? 1.5f : 0.0f) - 0.5f * a;
        T[idx] = t;
        Z[idx] = t;
    }
}

__global__ __launch_bounds__(256)
void diag_std_k(const float* __restrict__ S, float* __restrict__ stdv)
{
    int idx = blockIdx.x * blockDim.x + threadIdx.x;   // B*512
    if (idx < BATCH * CDIM) {
        int b = idx >> 9, i = idx & 511;
        float d = S[((size_t)b << 18) + ((size_t)i * CDIM) + i];
        d = fmaxf(d, EPSF);
        stdv[idx] = sqrtf(d);
    }
}

__global__ __launch_bounds__(256)
void corr_norm_k(float* __restrict__ S, const float* __restrict__ stdv, size_t total)
{
    for (size_t idx = (size_t)blockIdx.x * blockDim.x + threadIdx.x;
         idx < total; idx += (size_t)gridDim.x * blockDim.x) {
        int b = (int)(idx >> 18);
        int i = (int)((idx >> 9) & 511);
        int j = (int)(idx & 511);
        float nrm = stdv[b * CDIM + i] * stdv[b * CDIM + j];
        if (nrm == 0.0f) nrm = 1.0f;
        S[idx] = S[idx] / nrm;
    }
}

__global__ __launch_bounds__(256)
void sub_identity_k(float* __restrict__ X, size_t total)
{
    for (size_t idx = (size_t)blockIdx.x * blockDim.x + threadIdx.x;
         idx < total; idx += (size_t)gridDim.x * blockDim.x) {
        int i = (int)((idx >> 9) & 511);
        int j = (int)(idx & 511);
        if (i == j) X[idx] -= 1.0f;
    }
}

__global__ __launch_bounds__(256)
void init_scaled_identity_k(float* __restrict__ G, float c, size_t total)
{
    for (size_t idx = (size_t)blockIdx.x * blockDim.x + threadIdx.x;
         idx < total; idx += (size_t)gridDim.x * blockDim.x) {
        int i = (int)((idx >> 9) & 511);
        int j = (int)(idx & 511);
        G[idx] = (i == j) ? c : 0.0f;
    }
}

// zero diag, symmetrize, triu-vectorize
__global__ __launch_bounds__(256)
void finalize_triu_k(const float* __restrict__ logC, float* __restrict__ out)
{
    const int b = blockIdx.x >> 9;
    const int i = blockIdx.x & 511;
    const float* Lb = logC + ((size_t)b << 18);
    const int before = i * CDIM - (i * (i - 1)) / 2;   // elems before row i
    float* orow = out + (size_t)b * ((CDIM * (CDIM + 1)) / 2) + before;
    for (int j = i + threadIdx.x; j < CDIM; j += blockDim.x) {
        float v = (j == i) ? 0.0f
                           : (Lb[(size_t)i * CDIM + j] + Lb[(size_t)j * CDIM + i]);
        orow[j - i] = v;
    }
}

// ---------------------------------------------------------------------------
// host-side orchestration
// ---------------------------------------------------------------------------
static inline void launch_gemm(const float* A, const float* B, float* D,
                               const float* tr, float alpha, float beta,
                               int use_sqrt_tr, hipStream_t s)
{
    dim3 grid(CDIM / TN, CDIM / TM, BATCH);
    wmma_gemm_nn<<<grid, dim3(256), 0, s>>>(A, B, D, tr, alpha, beta, use_sqrt_tr);
}

// out = sqrtm(in) [Newton-Schulz, 5 iters, trace-normalized, *sqrt(trace)]
// 'out' may alias 'in'. s1..s5 are distinct scratch matrix slots (also != in/out).
static void run_sqrtm(const float* in, float* out,
                      float* s1, float* s2, float* s3, float* s4, float* s5,
                      float* tr, size_t total, int ew_blocks, hipStream_t st)
{
    trace_k<<<BATCH, 256, 0, st>>>(in, tr);
    prep_ns_k<<<ew_blocks, 256, 0, st>>>(in, tr, s1, s2, s3, total);  // A,T,Z
    launch_gemm(s1, s2, s4, nullptr, 1.0f, 0.0f, 0, st);              // Y = A@T

    float *Y = s4, *Z = s3, *f1 = s1, *f2 = s2, *f3 = s5;
    for (int it = 0; it < 3; ++it) {           // ITER_N - 2
        float* Tn = f1;
        launch_gemm(Z, Y, Tn, nullptr, -0.5f, 1.5f, 0, st);  // T = 1.5I - 0.5 Z@Y
        float* Yn = f2;
        launch_gemm(Y, Tn, Yn, nullptr, 1.0f, 0.0f, 0, st);  // Y = Y@T
        float* Zn = f3;
        launch_gemm(Tn, Z, Zn, nullptr, 1.0f, 0.0f, 0, st);  // Z = T@Z
        float *oY = Y, *oZ = Z, *oT = Tn;
        Y = Yn; Z = Zn; f1 = oY; f2 = oZ; f3 = oT;
    }
    float* U = f1;
    launch_gemm(Z, Y, U, nullptr, -0.5f, 1.5f, 0, st);       // U = 1.5I - 0.5 Z@Y
    launch_gemm(Y, U, out, tr, 1.0f, 0.0f, 1, st);           // out = Y@U * sqrt(tr)
}

extern "C" void kernel_launch(void* const* d_in, const int* in_sizes, int n_in,
                              void* d_out, int out_size, void* d_ws, size_t ws_size,
                              hipStream_t stream)
{
    (void)in_sizes; (void)n_in; (void)out_size; (void)ws_size;
    const float* x = (const float*)d_in[0];
    float* out = (float*)d_out;

    const size_t mat_elems = (size_t)BATCH << 18;     // B*512*512
    const int ew_blocks = 8192;

    // workspace layout
    float* ws   = (float*)d_ws;
    float* tr   = ws;                                 // 128 (pad 256)
    float* stdv = ws + 256;                           // 65536
    float* Xc   = ws + 256 + (size_t)BATCH * CDIM;    // B*512*196
    float* base = Xc + (size_t)BATCH * CDIM * MDIM;
    float* slot[6];
    for (int i = 0; i < 6; ++i) slot[i] = base + (size_t)i * mat_elems;

    // 1) center
    center_k<<<BATCH * CDIM, 64, 0, stream>>>(x, Xc);

    // 2) cov -> slot0
    {
        dim3 grid(CDIM / TILE_N, CDIM / TILE_M, BATCH);
        wmma_cov_nt<<<grid, dim3(256), 0, stream>>>(Xc, slot[0]);
    }

    // 3) S = sqrtm(cov) (in place in slot0)
    run_sqrtm(slot[0], slot[0], slot[1], slot[2], slot[3], slot[4], slot[5],
              tr, mat_elems, ew_blocks, stream);

    // 4) correlation normalization (in place): Cm in slot0
    diag_std_k<<<(BATCH * CDIM) / 256, 256, 0, stream>>>(slot[0], stdv);
    corr_norm_k<<<ew_blocks, 256, 0, stream>>>(slot[0], stdv, mat_elems);

    // 5) logm via inverse scaling-and-squaring: 4 repeated square roots
    for (int j = 0; j < 4; ++j) {
        run_sqrtm(slot[0], slot[0], slot[1], slot[2], slot[3], slot[4], slot[5],
                  tr, mat_elems, ew_blocks, stream);
    }
    // R = X - I (slot0)
    sub_identity_k<<<ew_blocks, 256, 0, stream>>>(slot[0], mat_elems);

    // Gregory series, Horner: log(I+R) ~ R*(c1 I + R*(c2 I + ... + R*(c8 I)))
    const float coef[8] = { 1.0f, -0.5f, 1.0f/3.0f, -0.25f,
                            0.2f, -1.0f/6.0f, 1.0f/7.0f, -0.125f };
    init_scaled_identity_k<<<ew_blocks, 256, 0, stream>>>(slot[1], coef[7], mat_elems);
    float* G  = slot[1];
    float* Gn = slot[2];
    for (int i = 6; i >= 0; --i) {
        launch_gemm(slot[0], G, Gn, nullptr, 1.0f, coef[i], 0, stream); // Gn = R@G + c_i I
        float* t = G; G = Gn; Gn = t;
    }
    // logC = 2^4 * R @ G  -> slot3
    launch_gemm(slot[0], G, slot[3], nullptr, 16.0f, 0.0f, 0, stream);

    // 6) finalize: zero diag, symmetrize, triu
    finalize_triu_k<<<BATCH * CDIM, 256, 0, stream>>>(slot[3], out);
}